// DWT_58162447122635
// MI455X (gfx1250) — compile-verified
//
#include <hip/hip_runtime.h>
#include <stdint.h>

// Toolchain arity detection for the TDM builtin:
// amdgpu-toolchain (therock headers) ships the TDM header and uses the 6-arg form;
// ROCm 7.2 uses the 5-arg form.
#if __has_include(<hip/amd_detail/amd_gfx1250_TDM.h>)
#define TDM_ARITY6 1
#else
#define TDM_ARITY6 0
#endif

typedef float v2f  __attribute__((ext_vector_type(2)));
typedef float v8f  __attribute__((ext_vector_type(8)));
typedef unsigned int u32x4 __attribute__((ext_vector_type(4)));
typedef int i32x4 __attribute__((ext_vector_type(4)));
typedef int i32x8 __attribute__((ext_vector_type(8)));

// Problem constants (reference setup is fixed).
constexpr int N_  = 4, C_ = 64, H_ = 512, W_ = 512;
constexpr int OH_ = (H_ + 3) / 2;   // 257
constexpr int OW_ = (W_ + 3) / 2;   // 257
constexpr int TOH = 64;             // output rows per block
constexpr int TOW = 32;             // output cols per block
constexpr int ITH = 2 * TOH + 2;    // 130 input rows needed
constexpr int ITW = 2 * TOW + 2;    // 66 input cols needed
constexpr int TS  = 33;             // padded row stride of intermediate (bank spread)

__device__ __forceinline__ int refl(int q, int n) {
  q = (q < 0) ? (-q - 1) : q;           // half-sample symmetric, left
  q = (q >= n) ? (2 * n - 1 - q) : q;   // right
  return q;
}

// db2 dec_lo reversed (the convolution taps): g[0..3]
__device__ __forceinline__ float wcoef(int d) {
  float r = 0.0f;
  r = (d == 0) ?  0.48296291314469025f : r;
  r = (d == 1) ?  0.8365163037378079f  : r;
  r = (d == 2) ?  0.22414386804185735f : r;
  r = (d == 3) ? -0.12940952255092145f : r;
  return r;
}

__global__ __launch_bounds__(256) void dwt_ll_fused(const float* __restrict__ x,
                                                    float* __restrict__ out)
{
  __shared__ float Xs[ITH * ITW];   // raw input tile (TDM target: contiguous 66-wide rows)
  __shared__ float Ts[ITH * TS];    // after horizontal pass (padded stride)

  const int tid   = threadIdx.x;
  const int tw    = blockIdx.x;     // output col tile
  const int th    = blockIdx.y;     // output row tile
  const int plane = blockIdx.z;     // n*C + c

  const int H0 = th * TOH, W0 = tw * TOW;
  const int gh0 = 2 * H0 - 2, gw0 = 2 * W0 - 2;
  const float* px = x + (size_t)plane * H_ * W_;

  const bool interior = (gh0 >= 0) && (gh0 + ITH <= H_) &&
                        (gw0 >= 0) && (gw0 + ITW <= W_);

  if (interior) {
    if (tid < 32) {
      // ---- Tensor Data Mover: DMA the 130x66 fp32 tile (row stride 512) into LDS ----
      const uint64_t ga  = (uint64_t)(const void*)(px + (size_t)gh0 * W_ + gw0);
      const unsigned lds = (unsigned)(uint64_t)(void*)Xs;  // low 32 bits = LDS byte offset

      u32x4 d0;
      d0.x = 1u;                                            // count=1, user descriptor
      d0.y = lds;                                           // lds_addr
      d0.z = (unsigned)(ga & 0xffffffffu);                  // global_addr[31:0]
      d0.w = (unsigned)((ga >> 32) & 0x1ffffffu) | (2u << 30); // addr[56:32] | type=2

      const uint64_t s0 = (uint64_t)W_;                     // tensor_dim0_stride (elems)
      const uint64_t s1 = (uint64_t)W_ * (uint64_t)H_;      // tensor_dim1_stride (2D: unused)
      i32x8 d1;
      d1[0] = (int)(2u << 16);                                               // data_size = 4B
      d1[1] = (int)(((unsigned)W_ & 0xffffu) << 16);                         // tensor_dim0 lo
      d1[2] = (int)((((unsigned)W_ >> 16) & 0xffffu) |
                    (((unsigned)H_ & 0xffffu) << 16));                       // td0 hi | td1 lo
      d1[3] = (int)((((unsigned)H_ >> 16) & 0xffffu) |
                    ((unsigned)ITW << 16));                                  // td1 hi | tile_dim0
      d1[4] = (int)((unsigned)ITH & 0xffffu);                                // tile_dim1 | tile_dim2=0
      d1[5] = (int)(unsigned)(s0 & 0xffffffffu);                             // stride0[31:0]
      d1[6] = (int)(((s0 >> 32) & 0xffffu) | ((s1 & 0xffffu) << 16));        // s0 hi | s1 lo
      d1[7] = (int)((s1 >> 16) & 0xffffffffu);                               // s1[47:16]
      i32x4 z4 = {0, 0, 0, 0};
#if TDM_ARITY6
      i32x8 z8 = {0, 0, 0, 0, 0, 0, 0, 0};
      __builtin_amdgcn_tensor_load_to_lds(d0, d1, z4, z4, z8, 0);
#else
      __builtin_amdgcn_tensor_load_to_lds(d0, d1, z4, z4, 0);
#endif
      __builtin_amdgcn_s_wait_tensorcnt((short)0);
    }
  } else {
    // Border tile: gather with symmetric reflection (every Xs element defined).
    for (int idx = tid; idx < ITH * ITW; idx += 256) {
      const int r = idx / ITW;
      const int c = idx - r * ITW;
      Xs[idx] = px[(size_t)refl(gh0 + r, H_) * W_ + refl(gw0 + c, W_)];
    }
  }
  __syncthreads();

  // Horizontal low-pass + decimate: Ts[r][wc] = sum_j g[j] * Xs[r][2*wc+j]
  for (int idx = tid; idx < ITH * TOW; idx += 256) {
    const int r  = idx >> 5;           // TOW == 32
    const int wc = idx & (TOW - 1);
    const float* row = &Xs[r * ITW + 2 * wc];
    Ts[r * TS + wc] = 0.48296291314469025f * row[0]
                    + 0.8365163037378079f  * row[1]
                    + 0.22414386804185735f * row[2]
                    - 0.12940952255092145f * row[3];
  }
  __syncthreads();

  // Vertical low-pass as banded matmul via V_WMMA_F32_16X16X4_F32.
  // Each wave owns one 16x16 output subtile: out[m][n] = sum_k A[m][k] * Ts[32*sub_r+k][16*sub_c+n],
  // A[m][k] = g[k-2m] inside the band (k in 0..33, padded to 36 = 9 chunks of K=4).
  const int wave  = tid >> 5;
  const int lane  = tid & 31;
  const int sub_r = wave >> 1;          // 0..3
  const int sub_c = wave & 1;           // 0..1
  const int half  = lane >> 4;          // upper-half lanes
  const int nn    = lane & 15;          // A: M row; B/C: N column

  v8f acc = {0.f, 0.f, 0.f, 0.f, 0.f, 0.f, 0.f, 0.f};
  const int tbase = 32 * sub_r;
  const int cbase = 16 * sub_c;

#pragma unroll
  for (int kc = 0; kc < 9; ++kc) {
    const int k0 = 4 * kc + 2 * half;                 // absolute K of this lane's pair
    v2f A = { wcoef(k0 - 2 * nn), wcoef(k0 + 1 - 2 * nn) };
    // Clamp K>=34 to 33: those columns have exactly-zero coefficients for every
    // valid row, and the clamped values are finite (no NaN/garbage reads).
    const int kr0 = (k0     < 34) ? k0     : 33;
    const int kr1 = (k0 + 1 < 34) ? k0 + 1 : 33;
    v2f B = { Ts[(tbase + kr0) * TS + cbase + nn],
              Ts[(tbase + kr1) * TS + cbase + nn] };
    acc = __builtin_amdgcn_wmma_f32_16x16x4_f32(false, A, false, B,
                                                (short)0, acc, false, false);
  }

  // Store: C/D layout — VGPR r: lanes 0-15 hold M=r, lanes 16-31 hold M=r+8; N = lane&15.
  const int wout  = W0 + cbase + nn;
  const int hbase = H0 + 16 * sub_r + 8 * half;
  if (wout < OW_) {
    float* pout = out + (size_t)plane * OH_ * OW_;
#pragma unroll
    for (int r = 0; r < 8; ++r) {
      const int hout = hbase + r;
      if (hout < OH_) pout[(size_t)hout * OW_ + wout] = acc[r];
    }
  }
}

extern "C" void kernel_launch(void* const* d_in, const int* in_sizes, int n_in,
                              void* d_out, int out_size, void* d_ws, size_t ws_size,
                              hipStream_t stream)
{
  (void)in_sizes; (void)n_in; (void)out_size; (void)d_ws; (void)ws_size;
  const float* x = (const float*)d_in[0];
  float* out = (float*)d_out;
  dim3 grid((OW_ + TOW - 1) / TOW,     // 9
            (OH_ + TOH - 1) / TOH,     // 5
            N_ * C_);                  // 256
  dwt_ll_fused<<<grid, dim3(256), 0, stream>>>(x, out);
}